// Correlation_34952443855265
// MI455X (gfx1250) — compile-verified
//
#include <hip/hip_runtime.h>

// CDNA5 / gfx1250, wave32.
// -mean(einsum('itj,itl->ijl', x, y)) collapses algebraically to
//   -1/(B*C*C) * sum_rows ( rowsum(x_row) * rowsum(y_row) )
// => memory-bound streaming reduction: 2 x 64 MB read once at 23.3 TB/s
//    (~5.8 us floor). No GEMM needed; the [B,C,C] intermediate is never formed.
// V_WMMA_F32_16X16X4_F32 (exact f32) is the accumulation engine for
// sum_t rx*ry, via half-symmetric rank-2 outer-product updates.

typedef float v2f __attribute__((ext_vector_type(2)));
typedef float v8f __attribute__((ext_vector_type(8)));

__device__ __forceinline__ float swizzle_xor(float v, int imm_offset_encoded);

// ds_swizzle_b32 group-of-32 mode: new_lane = ((lane & and) | or) ^ xor
//   offset = (xor << 10) | (or << 5) | and   (bit15 = 0)
#define SWZ_XOR16 0x401F
#define SWZ_XOR8  0x201F
#define SWZ_XOR4  0x101F
#define SWZ_XOR2  0x081F
#define SWZ_XOR1  0x041F

__device__ __forceinline__ float swz_f32_xor16(float v) {
    return __builtin_bit_cast(float,
        __builtin_amdgcn_ds_swizzle(__builtin_bit_cast(int, v), SWZ_XOR16));
}

__global__ void corr_zero_kernel(float* out) {
    if (threadIdx.x == 0) out[0] = 0.0f;
}

// Per wave, per row t:
//   px = lane-local partial sum of x[row] (32 coalesced elems per lane)
//   py = same for y[row], pyp = py from lane^16 (one ds_swizzle)
//
// A operand = (px, 0) in ALL lanes. Documented 16x4 f32 A layout
// (lanes 0-15: VGPR0=K0, VGPR1=K1; lanes 16-31: VGPR0=K2, VGPR1=K3) gives:
//   A[m][0] = px[m], A[m][2] = px[m+16], A[m][1] = A[m][3] = 0.
// Two chained WMMAs with B1 = (py, py) and B2 = (pyp, pyp):
// for ANY (vgpr,half)->K mapping of the 4x16 B layout (N = lane&15),
//   B1[k][n] + B2[k][n] = py[n] + py[n+16]   for every slot k,
// so (D1+D2)[m][n] += (px[m]+px[m+16]) * (py[n]+py[n+16]) and the sum over
// all 256 accumulator entries is exactly (sum_i px_i)(sum_j py_j) = rx*ry.
// The v8f C register carries the sum over t for the whole row loop ->
// a single cross-lane reduction per wave at the end.
__global__ __launch_bounds__(256, 1) void corr_rowsum_wmma_kernel(
        const float* __restrict__ x,
        const float* __restrict__ y,
        float* __restrict__ out,
        int nrows, float scale) {
    const int lane = threadIdx.x & 31;
    const int wid  = (blockIdx.x * blockDim.x + threadIdx.x) >> 5;
    const int nw   = (gridDim.x * blockDim.x) >> 5;

    v8f acc = {};  // 16x16 f32 WMMA accumulator (persistent over rows)

    for (int row = wid; row < nrows; row += nw) {  // wave-uniform: EXEC stays full
        const float4* __restrict__ xr =
            reinterpret_cast<const float4*>(x + (size_t)row * 1024u) + lane;
        const float4* __restrict__ yr =
            reinterpret_cast<const float4*>(y + (size_t)row * 1024u) + lane;

        // Coalesced streaming: lane L covers columns {4L + 128*i}, i=0..7.
        float4 vx = xr[0];
        float4 vy = yr[0];
        #pragma unroll
        for (int i = 1; i < 8; ++i) {
            float4 a = xr[i * 32];
            float4 b = yr[i * 32];
            vx.x += a.x; vx.y += a.y; vx.z += a.z; vx.w += a.w;
            vy.x += b.x; vy.y += b.y; vy.z += b.z; vy.w += b.w;
        }
        float px = (vx.x + vx.y) + (vx.z + vx.w);
        float py = (vy.x + vy.y) + (vy.z + vy.w);

        float pyp = swz_f32_xor16(py);       // single ds_swizzle, no clamp code

        v2f A;  A.x  = px;   A.y  = 0.0f;    // no cndmask, no px shuffle
        v2f B1; B1.x = py;   B1.y = py;
        v2f B2; B2.x = pyp;  B2.y = pyp;

        acc = __builtin_amdgcn_wmma_f32_16x16x4_f32(
                  false, A, false, B1, (short)0, acc, false, false);
        acc = __builtin_amdgcn_wmma_f32_16x16x4_f32(
                  false, A, false, B2, (short)0, acc, false, false);
    }

    // The 256 accumulator entries live exactly once across 8 VGPRs x 32 lanes:
    // lane-local sum, then a 5-level ds_swizzle xor tree (EXEC is all ones).
    float s = ((acc[0] + acc[1]) + (acc[2] + acc[3])) +
              ((acc[4] + acc[5]) + (acc[6] + acc[7]));
    s += __builtin_bit_cast(float,
         __builtin_amdgcn_ds_swizzle(__builtin_bit_cast(int, s), SWZ_XOR16));
    s += __builtin_bit_cast(float,
         __builtin_amdgcn_ds_swizzle(__builtin_bit_cast(int, s), SWZ_XOR8));
    s += __builtin_bit_cast(float,
         __builtin_amdgcn_ds_swizzle(__builtin_bit_cast(int, s), SWZ_XOR4));
    s += __builtin_bit_cast(float,
         __builtin_amdgcn_ds_swizzle(__builtin_bit_cast(int, s), SWZ_XOR2));
    s += __builtin_bit_cast(float,
         __builtin_amdgcn_ds_swizzle(__builtin_bit_cast(int, s), SWZ_XOR1));

    if (lane == 0)
        atomicAdd(out, s * scale);   // global_atomic_add_f32
}

extern "C" void kernel_launch(void* const* d_in, const int* in_sizes, int n_in,
                              void* d_out, int out_size, void* d_ws, size_t ws_size,
                              hipStream_t stream) {
    const float* x = (const float*)d_in[0];
    const float* y = (const float*)d_in[1];
    float* out = (float*)d_out;

    const int C = 1024;
    const int T = 2048;
    const int nrows = in_sizes[0] / C;              // B*T = 16384
    const int B = nrows / T;                        // 8
    const float scale = -1.0f / ((float)B * (float)C * (float)C);

    corr_zero_kernel<<<1, 32, 0, stream>>>(out);
    corr_rowsum_wmma_kernel<<<512, 256, 0, stream>>>(x, y, out, nrows, scale);
}